// GNNEncoder_53317724013285
// MI455X (gfx1250) — compile-verified
//
#include <hip/hip_runtime.h>
#include <hip/hip_bf16.h>

typedef float v2f __attribute__((ext_vector_type(2)));
typedef float v8f __attribute__((ext_vector_type(8)));

#define D_FEAT 128
#define N_GRAPHS 64

// ---------------------------------------------------------------------------
// Degree handling
// ---------------------------------------------------------------------------
__global__ __launch_bounds__(256) void k_deg_init(float* __restrict__ deg, int n) {
    int i = blockIdx.x * blockDim.x + threadIdx.x;
    if (i < n) deg[i] = 1.0f;  // self-loop
}

__global__ __launch_bounds__(256) void k_deg_count(const int* __restrict__ dst,
                                                   float* __restrict__ deg, int nEdges) {
    int e = blockIdx.x * blockDim.x + threadIdx.x;
    if (e < nEdges) atomicAdd(&deg[dst[e]], 1.0f);
}

__global__ __launch_bounds__(256) void k_rsqrt_inplace(float* __restrict__ deg, int n) {
    int i = blockIdx.x * blockDim.x + threadIdx.x;
    if (i < n) deg[i] = rsqrtf(deg[i]);  // deg >= 1 always (self-loop)
}

// ---------------------------------------------------------------------------
// FP32 WMMA GEMM:  out[n x 128] = A[n x 128] @ W[128 x 128]
// One wave per 16x16 output tile; K consumed 4 at a time via
// v_wmma_f32_16x16x4_f32. n must be a multiple of 16 (50000 = 3125*16).
//
// A 16x4 layout : lane(0-15): M=lane,   K={0,1}+koff ; lanes 16-31: K={2,3}
// B 4x16 layout : lane(0-15): N=lane,   K rows {0,1}+koff ; mirrored upper half
// C/D layout    : VGPR r <-> M = (lane>>4)*8 + r, N = lane&15
// ---------------------------------------------------------------------------
__global__ __launch_bounds__(256) void k_gemm_wmma(const float* __restrict__ A,
                                                   const float* __restrict__ W,
                                                   float* __restrict__ out,
                                                   int nRows) {
    const int lane    = threadIdx.x & 31;
    const int wave    = threadIdx.x >> 5;      // 8 waves -> 8 column tiles (128 cols)
    const int rowTile = blockIdx.x * 16;       // gridDim.x = nRows/16
    const int colTile = wave * 16;
    if (rowTile >= nRows) return;

    const int m    = lane & 15;                 // row within tile (A), col within tile (B)
    const int koff = (lane >> 4) * 2;           // which K pair this half-wave owns

    const float* aRow = A + (size_t)(rowTile + m) * D_FEAT;
    const int    col  = colTile + m;

    v8f acc = {0.f, 0.f, 0.f, 0.f, 0.f, 0.f, 0.f, 0.f};

#pragma unroll
    for (int k = 0; k < D_FEAT; k += 4) {
        v2f a, b;
        // contiguous pair A[row][k+koff], A[row][k+koff+1]
        a.x = aRow[k + koff];
        a.y = aRow[k + koff + 1];
        // B rows k+koff, k+koff+1 at column `col`
        b.x = W[(size_t)(k + koff)     * D_FEAT + col];
        b.y = W[(size_t)(k + koff + 1) * D_FEAT + col];
        acc = __builtin_amdgcn_wmma_f32_16x16x4_f32(
            /*neg_a=*/false, a, /*neg_b=*/false, b,
            /*c_mod=*/(short)0, acc, /*reuse_a=*/false, /*reuse_b=*/false);
    }

    const int mBase = (lane >> 4) * 8;
#pragma unroll
    for (int r = 0; r < 8; ++r) {
        out[(size_t)(rowTile + mBase + r) * D_FEAT + colTile + (lane & 15)] = acc[r];
    }
}

// ---------------------------------------------------------------------------
// Self-loop term + bias (also fully initializes `out`):
//   out[i][f] = xw[i][f] * dinv[i]^2 + b[f]
// ---------------------------------------------------------------------------
__global__ __launch_bounds__(256) void k_self_bias(const float* __restrict__ xw,
                                                   const float* __restrict__ dinv,
                                                   const float* __restrict__ bias,
                                                   float* __restrict__ out, int n) {
    long long idx = (long long)blockIdx.x * blockDim.x + threadIdx.x;
    long long tot = (long long)n * D_FEAT;
    if (idx < tot) {
        int i = (int)(idx >> 7);
        int f = (int)(idx & (D_FEAT - 1));
        float di = dinv[i];
        out[idx] = xw[idx] * di * di + bias[f];
    }
}

// ---------------------------------------------------------------------------
// Edge scatter-add: one wave32 per edge; lane moves 4 contiguous floats.
//   out[dst] += xw[src] * dinv[src] * dinv[dst]
// ---------------------------------------------------------------------------
__global__ __launch_bounds__(256) void k_edge_scatter(const float* __restrict__ xw,
                                                      const int* __restrict__ src,
                                                      const int* __restrict__ dst,
                                                      const float* __restrict__ dinv,
                                                      float* __restrict__ out, int nEdges) {
    long long tid  = (long long)blockIdx.x * blockDim.x + threadIdx.x;
    int       e    = (int)(tid >> 5);
    int       lane = (int)(tid & 31);
    if (e >= nEdges) return;

    int   s    = src[e];
    int   d    = dst[e];
    float norm = dinv[s] * dinv[d];

    const float4 v = *(const float4*)(xw + (size_t)s * D_FEAT + lane * 4);
    float* o = out + (size_t)d * D_FEAT + lane * 4;
    atomicAdd(o + 0, v.x * norm);
    atomicAdd(o + 1, v.y * norm);
    atomicAdd(o + 2, v.z * norm);
    atomicAdd(o + 3, v.w * norm);
}

__global__ __launch_bounds__(256) void k_relu(float* __restrict__ h, long long tot) {
    long long idx = (long long)blockIdx.x * blockDim.x + threadIdx.x;
    if (idx < tot) h[idx] = fmaxf(h[idx], 0.0f);
}

// ---------------------------------------------------------------------------
// Global mean pool
// ---------------------------------------------------------------------------
__global__ __launch_bounds__(256) void k_pool_zero(float* __restrict__ psum,
                                                   float* __restrict__ pcnt) {
    int i = blockIdx.x * blockDim.x + threadIdx.x;
    if (i < N_GRAPHS * D_FEAT) psum[i] = 0.0f;
    if (i < N_GRAPHS) pcnt[i] = 0.0f;
}

__global__ __launch_bounds__(256) void k_pool_acc(const float* __restrict__ h,
                                                  const int* __restrict__ batch,
                                                  float* __restrict__ psum,
                                                  float* __restrict__ pcnt, int n) {
    long long tid  = (long long)blockIdx.x * blockDim.x + threadIdx.x;
    int       i    = (int)(tid >> 5);
    int       lane = (int)(tid & 31);
    if (i >= n) return;
    int g = batch[i];
    const float4 v = *(const float4*)(h + (size_t)i * D_FEAT + lane * 4);
    float* o = psum + (size_t)g * D_FEAT + lane * 4;
    atomicAdd(o + 0, v.x);
    atomicAdd(o + 1, v.y);
    atomicAdd(o + 2, v.z);
    atomicAdd(o + 3, v.w);
    if (lane == 0) atomicAdd(&pcnt[g], 1.0f);
}

__global__ __launch_bounds__(256) void k_pool_fin(const float* __restrict__ psum,
                                                  const float* __restrict__ pcnt,
                                                  float* __restrict__ out) {
    int i = blockIdx.x * blockDim.x + threadIdx.x;
    if (i < N_GRAPHS * D_FEAT) {
        int g = i >> 7;
        out[i] = psum[i] / fmaxf(pcnt[g], 1.0f);
    }
}

// ---------------------------------------------------------------------------
// Launcher
// ---------------------------------------------------------------------------
extern "C" void kernel_launch(void* const* d_in, const int* in_sizes, int n_in,
                              void* d_out, int out_size, void* d_ws, size_t ws_size,
                              hipStream_t stream) {
    const float* x    = (const float*)d_in[0];   // [N,128]
    const int*   eidx = (const int*)d_in[1];     // [2,E] flat
    const int*   bat  = (const int*)d_in[2];     // [N]
    const float* W1   = (const float*)d_in[3];   // [128,128]
    const float* b1   = (const float*)d_in[4];   // [128]
    const float* W2   = (const float*)d_in[5];   // [128,128]
    const float* b2   = (const float*)d_in[6];   // [128]
    float*       out  = (float*)d_out;           // [64,128]

    const int N = in_sizes[0] / D_FEAT;          // 50000
    const int E = in_sizes[1] / 2;               // 800000
    const int* src = eidx;
    const int* dst = eidx + E;

    // Workspace layout (floats)
    float* ws   = (float*)d_ws;
    size_t off  = 0;
    float* dinv = ws + off; off += (size_t)((N + 63) / 64) * 64;
    float* buf1 = ws + off; off += (size_t)N * D_FEAT;      // xw / hw2
    float* buf2 = ws + off; off += (size_t)N * D_FEAT;      // h1 / h2
    float* psum = ws + off; off += N_GRAPHS * D_FEAT;
    float* pcnt = ws + off; off += N_GRAPHS;
    (void)ws_size; (void)n_in; (void)out_size;

    const int       TB       = 256;
    const long long totElems = (long long)N * D_FEAT;
    const int       gElem    = (int)((totElems + TB - 1) / TB);
    const int       gNode    = (N + TB - 1) / TB;
    const int       gEdgeT   = (E + TB - 1) / TB;                       // thread/edge
    const int       gEdgeW   = (int)(((long long)E * 32 + TB - 1) / TB); // wave/edge
    const int       gNodeW   = (int)(((long long)N * 32 + TB - 1) / TB); // wave/node
    const int       gTile    = N / 16;                                   // 3125

    // Degrees
    k_deg_init<<<gNode, TB, 0, stream>>>(dinv, N);
    k_deg_count<<<gEdgeT, TB, 0, stream>>>(dst, dinv, E);
    k_rsqrt_inplace<<<gNode, TB, 0, stream>>>(dinv, N);

    // ---- Layer 1 ----
    k_gemm_wmma<<<gTile, TB, 0, stream>>>(x, W1, buf1, N);               // buf1 = X@W1
    k_self_bias<<<gElem, TB, 0, stream>>>(buf1, dinv, b1, buf2, N);      // buf2 = self + b1
    k_edge_scatter<<<gEdgeW, TB, 0, stream>>>(buf1, src, dst, dinv, buf2, E);
    k_relu<<<gElem, TB, 0, stream>>>(buf2, totElems);                    // buf2 = h1

    // ---- Layer 2 ----
    k_gemm_wmma<<<gTile, TB, 0, stream>>>(buf2, W2, buf1, N);            // buf1 = H1@W2
    k_self_bias<<<gElem, TB, 0, stream>>>(buf1, dinv, b2, buf2, N);      // buf2 = self + b2
    k_edge_scatter<<<gEdgeW, TB, 0, stream>>>(buf1, src, dst, dinv, buf2, E);

    // ---- Mean pool ----
    k_pool_zero<<<(N_GRAPHS * D_FEAT + TB - 1) / TB, TB, 0, stream>>>(psum, pcnt);
    k_pool_acc<<<gNodeW, TB, 0, stream>>>(buf2, bat, psum, pcnt, N);
    k_pool_fin<<<(N_GRAPHS * D_FEAT + TB - 1) / TB, TB, 0, stream>>>(psum, pcnt, out);
}